// RPNLayer_11158325035064
// MI455X (gfx1250) — compile-verified
//
#include <hip/hip_runtime.h>
#include <hip/hip_bf16.h>

typedef float v2f __attribute__((ext_vector_type(2)));
typedef float v4f __attribute__((ext_vector_type(4)));
typedef float v8f __attribute__((ext_vector_type(8)));

#define ANCHOR_NUM 8
#define IN_DIM 1024
#define NLOGIT 16          // ANCHOR_NUM * CLASS_NUM
#define WAVES_PER_BLOCK 8
#define ROWS_PER_TILE 16
#define KCHUNK 32
#define APAD 36            // LDS row stride (floats): 36=4*9, 9 coprime 16 -> conflict-free b64 reads
#define NPRE 4             // float4 staging regs per lane per chunk (16*32/(32*4))

// ---------------------------------------------------------------------------
// Main kernel: GEMM via V_WMMA_F32_16X16X4_F32, double-buffered LDS staging,
// dual accumulators, fused CE / argmax epilogue.
// One wave32 per 16-row tile; 8 tiles per 256-thread block.
// ---------------------------------------------------------------------------
__global__ __launch_bounds__(256) void rpn_main(
    const float* __restrict__ x,      // (rows, 1024)
    const float* __restrict__ W,      // (1024, 16) k-major
    const float* __restrict__ bias,   // (16,)
    const int*   __restrict__ labels, // (rows, 8) in {-1,0,1}
    float* __restrict__ out_pred,     // (rows, 8)
    float* __restrict__ out_mask,     // (rows, 8)
    float* __restrict__ ws)           // per-block (sum, count)
{
    __shared__ __align__(16) float As[WAVES_PER_BLOCK][2][ROWS_PER_TILE * APAD];
    __shared__ float sred[WAVES_PER_BLOCK * 2];

    const int lane = threadIdx.x & 31;
    const int wave = threadIdx.x >> 5;
    const int half = lane >> 4;       // which 16-lane half
    const int l15  = lane & 15;

    const int tileBase = (blockIdx.x * WAVES_PER_BLOCK + wave) * ROWS_PER_TILE;

    // Per-lane staging coordinates: chunk c = it*32 + lane; 8 float4 per row.
    const int srow0   = lane >> 3;          // rows advance by 4 per it
    const int schunk  = lane & 7;
    const float* gbase = x + (size_t)(tileBase + srow0) * IN_DIM + schunk * 4;

    v8f acc0 = {}, acc1 = {};
    v4f pre[NPRE];

    // ---- Prologue: stream chunk 0 (NT: x is read exactly once) ----
    #pragma unroll
    for (int it = 0; it < NPRE; ++it)
        pre[it] = __builtin_nontemporal_load(
            (const v4f*)(gbase + (size_t)(it * 4) * IN_DIM));
    #pragma unroll
    for (int it = 0; it < NPRE; ++it)
        *(v4f*)&As[wave][0][(srow0 + it * 4) * APAD + schunk * 4] = pre[it];

    for (int k0 = 0; k0 < IN_DIM; k0 += KCHUNK) {
        const int buf  = (k0 / KCHUNK) & 1;
        const bool more = (k0 + KCHUNK) < IN_DIM;

        // ---- Issue next chunk's global loads BEFORE computing (overlap) ----
        if (more) {
            #pragma unroll
            for (int it = 0; it < NPRE; ++it)
                pre[it] = __builtin_nontemporal_load(
                    (const v4f*)(gbase + k0 + KCHUNK + (size_t)(it * 4) * IN_DIM));
        }

        // ---- 8 WMMAs over this K-chunk, alternating accumulators ----
        #pragma unroll
        for (int kk = 0; kk < KCHUNK; kk += 8) {
            {
                const int k = k0 + kk;
                // A 16x4 f32: lanes 0-15 (M=lane, K=kk..kk+1); lanes 16-31 (K=kk+2..kk+3)
                v2f a = *(const v2f*)&As[wave][buf][l15 * APAD + kk + 2 * half];
                v2f bm;   // B 4x16 f32: vgpr0 = row k|k+2, vgpr1 = row k+1|k+3
                bm.x = W[(size_t)(k     + 2 * half) * NLOGIT + l15];
                bm.y = W[(size_t)(k + 1 + 2 * half) * NLOGIT + l15];
                acc0 = __builtin_amdgcn_wmma_f32_16x16x4_f32(
                    false, a, false, bm, (short)0, acc0, false, false);
            }
            {
                const int k = k0 + kk + 4;
                v2f a = *(const v2f*)&As[wave][buf][l15 * APAD + kk + 4 + 2 * half];
                v2f bm;
                bm.x = W[(size_t)(k     + 2 * half) * NLOGIT + l15];
                bm.y = W[(size_t)(k + 1 + 2 * half) * NLOGIT + l15];
                acc1 = __builtin_amdgcn_wmma_f32_16x16x4_f32(
                    false, a, false, bm, (short)0, acc1, false, false);
            }
        }

        // ---- Drain next chunk into the other LDS buffer (after compute) ----
        if (more) {
            #pragma unroll
            for (int it = 0; it < NPRE; ++it)
                *(v4f*)&As[wave][buf ^ 1][(srow0 + it * 4) * APAD + schunk * 4] = pre[it];
        }
    }

    // ---- Epilogue: merge accumulators, bias add, spill logits to LDS ----
    const float bv = bias[l15];       // N = l15 for both lane halves
    #pragma unroll
    for (int r = 0; r < 8; ++r) {
        const int M = r + 8 * half;   // C/D layout: VGPR r -> M=r (lo half), M=r+8 (hi half)
        As[wave][0][M * NLOGIT + l15] = acc0[r] + acc1[r] + bv;
    }

    // ---- Per-anchor: argmax, NLL, masks. 16 rows * 8 anchors = 4 iters ----
    float lsum = 0.f, lcnt = 0.f;
    #pragma unroll
    for (int t = 0; t < 4; ++t) {
        const int item = t * 32 + lane;
        const int row  = item >> 3;
        const int a    = item & 7;
        const float l0 = As[wave][0][row * NLOGIT + a * 2];
        const float l1 = As[wave][0][row * NLOGIT + a * 2 + 1];
        const int   g  = tileBase + row;
        const int  lab = labels[(size_t)g * ANCHOR_NUM + a];
        const int pred = (l1 > l0) ? 1 : 0;          // ties -> 0, matches argmax
        const float m   = fmaxf(l0, l1);
        const float lse = m + __logf(__expf(l0 - m) + __expf(l1 - m));
        const float nll = lse - ((lab > 0) ? l1 : l0);
        if (lab != -1) { lsum += nll; lcnt += 1.f; }
        out_pred[(size_t)g * ANCHOR_NUM + a] = (float)pred;
        out_mask[(size_t)g * ANCHOR_NUM + a] = (pred == 1 && lab != -1) ? 1.f : 0.f;
    }

    // ---- Deterministic block-level loss partial ----
    for (int off = 16; off > 0; off >>= 1) {
        lsum += __shfl_down(lsum, off);
        lcnt += __shfl_down(lcnt, off);
    }
    if (lane == 0) { sred[wave * 2] = lsum; sred[wave * 2 + 1] = lcnt; }
    __syncthreads();
    if (threadIdx.x == 0) {
        float s = 0.f, c = 0.f;
        #pragma unroll
        for (int w = 0; w < WAVES_PER_BLOCK; ++w) { s += sred[2 * w]; c += sred[2 * w + 1]; }
        ws[blockIdx.x * 2]     = s;
        ws[blockIdx.x * 2 + 1] = c;
    }
}

// ---------------------------------------------------------------------------
// Final deterministic reduction of per-block partials -> loss scalar.
// ---------------------------------------------------------------------------
__global__ __launch_bounds__(256) void rpn_reduce(const float* __restrict__ ws,
                                                  int nblk, float* __restrict__ out0)
{
    __shared__ float ss[256], sc[256];
    const int t = threadIdx.x;
    float s = 0.f, c = 0.f;
    for (int i = t; i < nblk; i += 256) { s += ws[2 * i]; c += ws[2 * i + 1]; }
    ss[t] = s; sc[t] = c;
    __syncthreads();
    for (int off = 128; off > 0; off >>= 1) {
        if (t < off) { ss[t] += ss[t + off]; sc[t] += sc[t + off]; }
        __syncthreads();
    }
    if (t == 0) out0[0] = ss[0] / fmaxf(sc[0], 1.0f);
}

extern "C" void kernel_launch(void* const* d_in, const int* in_sizes, int n_in,
                              void* d_out, int out_size, void* d_ws, size_t ws_size,
                              hipStream_t stream) {
    const float* x      = (const float*)d_in[0];
    const float* W      = (const float*)d_in[1];
    const float* b      = (const float*)d_in[2];
    const int*   labels = (const int*)d_in[3];

    const int rows  = in_sizes[0] / IN_DIM;   // 32768
    const int nAnch = in_sizes[3];            // rows * ANCHOR_NUM = 262144

    float* out      = (float*)d_out;
    float* out_pred = out + 1;
    float* out_mask = out + 1 + nAnch;
    float* ws       = (float*)d_ws;

    const int tiles  = rows / ROWS_PER_TILE;            // 2048
    const int blocks = tiles / WAVES_PER_BLOCK;         // 256

    rpn_main<<<blocks, 256, 0, stream>>>(x, W, b, labels, out_pred, out_mask, ws);
    rpn_reduce<<<1, 256, 0, stream>>>(ws, blocks, out);
}